// GeoSAGEConv_31894427140226
// MI455X (gfx1250) — compile-verified
//
#include <hip/hip_runtime.h>
#include <hip/hip_bf16.h>
#include <math.h>

typedef __attribute__((ext_vector_type(2))) float v2f;
typedef __attribute__((ext_vector_type(8))) float v8f;

#define FEAT 64
#define WPAD 66   // LDS row stride (floats) to break bank conflicts

// ---------------------------------------------------------------------------
// zero fill
// ---------------------------------------------------------------------------
__global__ void zero_f32(float* __restrict__ p, int n) {
    int i = blockIdx.x * blockDim.x + threadIdx.x;
    int stride = gridDim.x * blockDim.x;
    for (; i < n; i += stride) p[i] = 0.0f;
}

// ---------------------------------------------------------------------------
// Edge aggregation: one wave (32 lanes) per edge, 2 features per lane.
//   agg[dst] += x[src];  cnt[dst] += 1 (layer-1 pass only)
// ---------------------------------------------------------------------------
__global__ __launch_bounds__(256) void edge_agg(
    const float* __restrict__ x, const int* __restrict__ src,
    const int* __restrict__ dst, float* __restrict__ agg,
    float* __restrict__ cnt, int E, int addCnt)
{
    int t = blockIdx.x * blockDim.x + threadIdx.x;
    int e = t >> 5;
    int lane = t & 31;
    if (e >= E) return;
    int s = src[e];
    int d = dst[e];
    const v2f v = *(const v2f*)(x + (size_t)s * FEAT + lane * 2);
    atomicAdd(&agg[(size_t)d * FEAT + lane * 2 + 0], v.x);
    atomicAdd(&agg[(size_t)d * FEAT + lane * 2 + 1], v.y);
    if (addCnt && lane == 0) atomicAdd(&cnt[d], 1.0f);
}

// ---------------------------------------------------------------------------
// Layer 1: out = relu(L2norm(mean @ Wl.T + b + x @ Wr.T))   (64 -> 64)
// One wave per 16-node tile. WMMA f32 16x16x4, K=64 (16 steps), 4 N-tiles.
// ---------------------------------------------------------------------------
__global__ __launch_bounds__(256) void sage_layer1(
    const float* __restrict__ agg, const float* __restrict__ cnt,
    const float* __restrict__ x,
    const float* __restrict__ Wl, const float* __restrict__ b,
    const float* __restrict__ Wr,
    float* __restrict__ hout, int numTiles)
{
    __shared__ float sWl[64 * WPAD];
    __shared__ float sWr[64 * WPAD];
    for (int i = threadIdx.x; i < 64 * 64; i += 256) {
        int n = i >> 6, k = i & 63;
        sWl[n * WPAD + k] = Wl[i];
        sWr[n * WPAD + k] = Wr[i];
    }
    __syncthreads();

    int wave = threadIdx.x >> 5;
    int lane = threadIdx.x & 31;
    int tile = blockIdx.x * 8 + wave;
    if (tile >= numTiles) return;

    int nlo  = lane & 15;       // row m for A frag / col n for B frag
    int lh   = lane >> 4;       // lane half selects K pair
    int node = tile * 16 + nlo;

    float invc = 1.0f / fmaxf(cnt[node], 1.0f);
    const float* aggRow = agg + (size_t)node * FEAT;
    const float* xRow   = x   + (size_t)node * FEAT;

    v8f acc[4] = {};
#pragma unroll
    for (int kb = 0; kb < 16; ++kb) {
        int k0 = kb * 4 + lh * 2;
        v2f am, ax;
        am.x = aggRow[k0] * invc;  am.y = aggRow[k0 + 1] * invc;
        ax.x = xRow[k0];           ax.y = xRow[k0 + 1];
#pragma unroll
        for (int nt = 0; nt < 4; ++nt) {
            int n = nt * 16 + nlo;
            v2f bl = *(const v2f*)&sWl[n * WPAD + k0];
            v2f br = *(const v2f*)&sWr[n * WPAD + k0];
            acc[nt] = __builtin_amdgcn_wmma_f32_16x16x4_f32(
                false, am, false, bl, (short)0, acc[nt], false, false);
            acc[nt] = __builtin_amdgcn_wmma_f32_16x16x4_f32(
                false, ax, false, br, (short)0, acc[nt], false, false);
        }
    }

    // bias
#pragma unroll
    for (int nt = 0; nt < 4; ++nt) {
        float bias = b[nt * 16 + nlo];
#pragma unroll
        for (int v = 0; v < 8; ++v) acc[nt][v] += bias;
    }

    // per-row L2 normalize + relu + store.  Row m = v + 8*lh, n = nt*16 + nlo.
#pragma unroll
    for (int v = 0; v < 8; ++v) {
        float s = 0.0f;
#pragma unroll
        for (int nt = 0; nt < 4; ++nt) { float t = acc[nt][v]; s += t * t; }
        s += __shfl_xor(s, 1);
        s += __shfl_xor(s, 2);
        s += __shfl_xor(s, 4);
        s += __shfl_xor(s, 8);
        float inv = 1.0f / fmaxf(sqrtf(s), 1e-12f);
        int row = tile * 16 + v + 8 * lh;
        float* orow = hout + (size_t)row * FEAT;
#pragma unroll
        for (int nt = 0; nt < 4; ++nt)
            orow[nt * 16 + nlo] = fmaxf(acc[nt][v] * inv, 0.0f);
    }
}

// ---------------------------------------------------------------------------
// Layer 2: out = log_softmax(mean @ Wl.T + b + h @ Wr.T)   (64 -> 10)
// N padded to 16 with zero weight columns; 1 N-tile, K=64.
// ---------------------------------------------------------------------------
__global__ __launch_bounds__(256) void sage_layer2(
    const float* __restrict__ agg, const float* __restrict__ cnt,
    const float* __restrict__ h,
    const float* __restrict__ Wl, const float* __restrict__ b,
    const float* __restrict__ Wr,
    float* __restrict__ out, int numTiles)
{
    __shared__ float sWl[16 * WPAD];
    __shared__ float sWr[16 * WPAD];
    for (int i = threadIdx.x; i < 16 * WPAD; i += 256) { sWl[i] = 0.0f; sWr[i] = 0.0f; }
    __syncthreads();
    for (int i = threadIdx.x; i < 10 * 64; i += 256) {
        int n = i >> 6, k = i & 63;
        sWl[n * WPAD + k] = Wl[i];
        sWr[n * WPAD + k] = Wr[i];
    }
    __syncthreads();

    int wave = threadIdx.x >> 5;
    int lane = threadIdx.x & 31;
    int tile = blockIdx.x * 8 + wave;
    if (tile >= numTiles) return;

    int nlo  = lane & 15;
    int lh   = lane >> 4;
    int node = tile * 16 + nlo;

    float invc = 1.0f / fmaxf(cnt[node], 1.0f);
    const float* aggRow = agg + (size_t)node * FEAT;
    const float* hRow   = h   + (size_t)node * FEAT;

    v8f acc = {};
#pragma unroll
    for (int kb = 0; kb < 16; ++kb) {
        int k0 = kb * 4 + lh * 2;
        v2f am, ah;
        am.x = aggRow[k0] * invc;  am.y = aggRow[k0 + 1] * invc;
        ah.x = hRow[k0];           ah.y = hRow[k0 + 1];
        v2f bl = *(const v2f*)&sWl[nlo * WPAD + k0];
        v2f br = *(const v2f*)&sWr[nlo * WPAD + k0];
        acc = __builtin_amdgcn_wmma_f32_16x16x4_f32(
            false, am, false, bl, (short)0, acc, false, false);
        acc = __builtin_amdgcn_wmma_f32_16x16x4_f32(
            false, ah, false, br, (short)0, acc, false, false);
    }

    float bias = (nlo < 10) ? b[nlo] : 0.0f;
#pragma unroll
    for (int v = 0; v < 8; ++v) acc[v] += bias;

    // log_softmax over the 10 valid columns of each row (row m = v + 8*lh)
#pragma unroll
    for (int v = 0; v < 8; ++v) {
        float val = acc[v];
        float mx = (nlo < 10) ? val : -3.0e38f;
        mx = fmaxf(mx, __shfl_xor(mx, 1));
        mx = fmaxf(mx, __shfl_xor(mx, 2));
        mx = fmaxf(mx, __shfl_xor(mx, 4));
        mx = fmaxf(mx, __shfl_xor(mx, 8));
        float ex = (nlo < 10) ? __expf(val - mx) : 0.0f;
        float se = ex;
        se += __shfl_xor(se, 1);
        se += __shfl_xor(se, 2);
        se += __shfl_xor(se, 4);
        se += __shfl_xor(se, 8);
        float res = val - mx - __logf(se);
        if (nlo < 10) {
            int row = tile * 16 + v + 8 * lh;
            out[(size_t)row * 10 + nlo] = res;
        }
    }
}

// ---------------------------------------------------------------------------
// Launch
// ---------------------------------------------------------------------------
extern "C" void kernel_launch(void* const* d_in, const int* in_sizes, int n_in,
                              void* d_out, int out_size, void* d_ws, size_t ws_size,
                              hipStream_t stream) {
    const float* features = (const float*)d_in[0];
    const int*   edges    = (const int*)d_in[1];
    const float* W1l      = (const float*)d_in[2];
    const float* b1       = (const float*)d_in[3];
    const float* W1r      = (const float*)d_in[4];
    const float* W2l      = (const float*)d_in[5];
    const float* b2       = (const float*)d_in[6];
    const float* W2r      = (const float*)d_in[7];

    const int N = in_sizes[0] / FEAT;          // 50000
    const int E = in_sizes[1] / 2;             // 1250000
    const int* src = edges;
    const int* dst = edges + E;

    // workspace layout: agg[N*64] | cnt[N] | h[N*64]
    float* agg = (float*)d_ws;
    float* cnt = agg + (size_t)N * FEAT;
    float* h   = cnt + (size_t)N;

    const int numTiles   = (N + 15) / 16;
    const int tileBlocks = (numTiles + 7) / 8;
    const int edgeBlocks = (int)(((long long)E * 32 + 255) / 256);

    // 1) zero agg + cnt (contiguous N*65 floats)
    zero_f32<<<2048, 256, 0, stream>>>(agg, N * (FEAT + 1));
    // 2) layer-1 aggregation
    edge_agg<<<edgeBlocks, 256, 0, stream>>>(features, src, dst, agg, cnt, E, 1);
    // 3) layer-1 linear + L2 normalize + relu -> h
    sage_layer1<<<tileBlocks, 256, 0, stream>>>(agg, cnt, features, W1l, b1, W1r,
                                                h, numTiles);
    // 4) zero agg
    zero_f32<<<2048, 256, 0, stream>>>(agg, N * FEAT);
    // 5) layer-2 aggregation on h
    edge_agg<<<edgeBlocks, 256, 0, stream>>>(h, src, dst, agg, cnt, E, 0);
    // 6) layer-2 linear + log_softmax -> d_out (N x 10, f32)
    sage_layer2<<<tileBlocks, 256, 0, stream>>>(agg, cnt, h, W2l, b2, W2r,
                                                (float*)d_out, numTiles);
}